// HighOrderActivation_83502754168910
// MI455X (gfx1250) — compile-verified
//
#include <hip/hip_runtime.h>

// HighOrderActivation: simplex interpolation over 2^K-entry tables.
// B=1024, D=512, K=4, O=64.  out[b,d,o] = sum_j coef[j] * params[d, ind[j], o]
// Recast per (d, 16-batch tile) as W[16x16] @ params_d[16x64] with
// V_WMMA_F32_16X16X4_F32 chained over K (4 chunks of 4).

#define B_BATCH 1024
#define D_DIM   512
#define K_ARITY 4
#define O_DIM   64
#define T_TBL   16

typedef __attribute__((ext_vector_type(2))) float v2f;
typedef __attribute__((ext_vector_type(8))) float v8f;

// compare-and-swap tracking original indices (strict < -> stable for ties)
__device__ __forceinline__ void cswap(float& va, int& ia, float& vb, int& ib) {
    const bool sw = vb < va;
    const float vlo = sw ? vb : va;
    const float vhi = sw ? va : vb;
    const int   ilo = sw ? ib : ia;
    const int   ihi = sw ? ia : ib;
    va = vlo; vb = vhi; ia = ilo; ib = ihi;
}

// W[m][c] = sum_j coef[j] * (ind[j] == c); ind values are distinct
__device__ __forceinline__ float wsel(int c, int t0, int t1, int t2, int t3,
                                      float c0, float c1, float c2, float c3) {
    float w = (t0 == c) ? c0 : 0.0f;
    w += (t1 == c) ? c1 : 0.0f;
    w += (t2 == c) ? c2 : 0.0f;
    w += (t3 == c) ? c3 : 0.0f;
    return w;
}

__global__ __launch_bounds__(128) void
HighOrderActivation_83502754168910_kernel(const float* __restrict__ X,
                                          const float* __restrict__ P,
                                          float* __restrict__ out) {
    const int lane = threadIdx.x & 31;
    const int wave = threadIdx.x >> 5;
    const int d    = blockIdx.y;
    const int b0   = blockIdx.x * 64 + wave * 16;   // 16-batch tile base for this wave

    const int m  = lane & 15;   // row (M) / col (N) position within 16-lane half
    const int hi = lane >> 4;   // which half of the wave

    // ---- per-lane simplex coefficients for batch row b = b0 + m ----
    const int b = b0 + m;
    const float4 xv = *reinterpret_cast<const float4*>(
        X + ((size_t)b * D_DIM + d) * K_ARITY);

    float v0 = xv.x, v1 = xv.y, v2 = xv.z, v3 = xv.w;
    int   i0 = 0,    i1 = 1,    i2 = 2,    i3 = 3;
    // 5-comparator sorting network for 4 elements (ascending)
    cswap(v0, i0, v1, i1);
    cswap(v2, i2, v3, i3);
    cswap(v0, i0, v2, i2);
    cswap(v1, i1, v3, i3);
    cswap(v1, i1, v2, i2);

    const float c0 = v0;
    const float c1 = v1 - v0;
    const float c2 = v2 - v1;
    const float c3 = v3 - v2;
    // reverse cumsum of bit masks -> corner table indices (distinct, in [1,15])
    const int t3 = 1 << i3;
    const int t2 = t3 + (1 << i2);
    const int t1 = t2 + (1 << i1);
    const int t0 = 15;  // all bits set

    // ---- A fragments: 16x4 f32 layout, lane holds W[m][4*kc + 2*hi + {0,1}] ----
    v2f a[4];
#pragma unroll
    for (int kc = 0; kc < 4; ++kc) {
        const int cb = 4 * kc + 2 * hi;
        a[kc].x = wsel(cb,     t0, t1, t2, t3, c0, c1, c2, c3);
        a[kc].y = wsel(cb + 1, t0, t1, t2, t3, c0, c1, c2, c3);
    }

    // ---- B fragments + WMMA chain + store, per 16-wide O tile ----
    const float* __restrict__ Pd = P + (size_t)d * T_TBL * O_DIM;

#pragma unroll
    for (int nt = 0; nt < 4; ++nt) {
        const int n0 = nt * 16;
        v8f acc = {};
#pragma unroll
        for (int kc = 0; kc < 4; ++kc) {
            // 4x16 f32 B layout: lane holds P[4*kc + 2*hi + r][n0 + m], r=0,1
            const int trow = 4 * kc + 2 * hi;
            v2f bf;
            bf.x = Pd[(size_t)trow * O_DIM + n0 + m];
            bf.y = Pd[(size_t)(trow + 1) * O_DIM + n0 + m];
            acc = __builtin_amdgcn_wmma_f32_16x16x4_f32(
                /*neg_a=*/false, a[kc], /*neg_b=*/false, bf,
                /*c_mod=*/(short)0, acc, /*reuse_a=*/false, /*reuse_b=*/false);
        }
        // C/D layout: VGPR r <-> M = r + 8*hi, N = m
#pragma unroll
        for (int r = 0; r < 8; ++r) {
            const int brow = b0 + 8 * hi + r;
            out[((size_t)brow * D_DIM + d) * O_DIM + n0 + m] = acc[r];
        }
    }
}

extern "C" void kernel_launch(void* const* d_in, const int* in_sizes, int n_in,
                              void* d_out, int out_size, void* d_ws, size_t ws_size,
                              hipStream_t stream) {
    const float* X = (const float*)d_in[0];   // [B, D, K] f32
    const float* P = (const float*)d_in[1];   // [D, 16, O] f32
    float* out = (float*)d_out;               // [B, D, O] f32

    dim3 grid(B_BATCH / 64, D_DIM, 1);        // 16 x 512 blocks
    dim3 block(128, 1, 1);                    // 4 waves; each wave: one 16-batch tile
    HighOrderActivation_83502754168910_kernel<<<grid, block, 0, stream>>>(X, P, out);
}